// GaLiTeAttentionLayer_64931315581522
// MI455X (gfx1250) — compile-verified
//
#include <hip/hip_runtime.h>
#include <math.h>

// ---------------- problem constants (from reference) ----------------
#define T_STEPS 64
#define BATCH   16
#define DIM     512
#define HEADS   8
#define DH      64
#define ETA_    4
#define FD      256            // ETA * DH
#define ROWS    (T_STEPS * BATCH)   // 1024
#define EPS_    1e-6f

typedef __attribute__((ext_vector_type(2))) float v2f;
typedef __attribute__((ext_vector_type(8))) float v8f;

__device__ __forceinline__ float sigmoidf_(float x) {
    return 1.0f / (1.0f + __expf(-x));
}

// =====================================================================
// GEMM: C[M,N] = X[M,K] * W[N,K]^T (+ bias), fp32 via V_WMMA_F32_16X16X4_F32
// Each wave computes a 32x32 tile of C: 2 A-fragments x 2 B-fragments ->
// 4 WMMAs per 4 fragment loads (2x the flops/byte of a 16x16 tile) with
// 4 independent accumulator chains. The k-loop is software-pipelined:
// next-step fragments are loaded before the current WMMAs consume theirs,
// so waits only cover loads issued a full iteration earlier.
// A layout: lane<16 -> A[M=lane][K=0..1] (float2); lane>=16 -> K=2..3
// B layout: lane<16 -> B[K=0..1][N=lane]; lane>=16 -> K=2..3 (N striped)
// C layout: VGPR j, lane L -> row = j + 8*(L>>4), col = L&15
// =====================================================================
__global__ void __launch_bounds__(256)
gemm_xwT_wmma(const float* __restrict__ X, const float* __restrict__ W,
              const float* __restrict__ bias, float* __restrict__ C,
              int M, int N, int K) {
    const int lane   = threadIdx.x & 31;
    const int waveIb = threadIdx.x >> 5;
    const int wave   = blockIdx.x * (blockDim.x >> 5) + waveIb;

    const int mTiles  = M >> 5;   // 32-row tiles
    const int nTiles  = N >> 5;   // 32-col tiles
    if (wave >= mTiles * nTiles) return;   // whole-wave exit; survivors EXEC==all 1s

    const int m0 = (wave % mTiles) << 5;
    const int n0 = (wave / mTiles) << 5;

    const int ln = lane & 15;
    const int kh = lane >> 4;     // K-half select

    const float* aRow0 = X + (size_t)(m0 + ln) * K + 2 * kh;
    const float* aRow1 = X + (size_t)(m0 + 16 + ln) * K + 2 * kh;
    const float* bRow0 = W + (size_t)(n0 + ln) * K + 2 * kh;
    const float* bRow1 = W + (size_t)(n0 + 16 + ln) * K + 2 * kh;

    v8f acc00 = {}, acc01 = {}, acc10 = {}, acc11 = {};

    // prologue: fragments for k=0
    v2f a0 = *(const v2f*)(aRow0);
    v2f a1 = *(const v2f*)(aRow1);
    v2f b0 = *(const v2f*)(bRow0);
    v2f b1 = *(const v2f*)(bRow1);

#pragma unroll 2
    for (int k0 = 4; k0 < K; k0 += 4) {
        // issue next-step loads first (software pipeline)
        v2f na0 = *(const v2f*)(aRow0 + k0);
        v2f na1 = *(const v2f*)(aRow1 + k0);
        v2f nb0 = *(const v2f*)(bRow0 + k0);
        v2f nb1 = *(const v2f*)(bRow1 + k0);

        acc00 = __builtin_amdgcn_wmma_f32_16x16x4_f32(
                    false, a0, false, b0, (short)0, acc00, false, false);
        acc01 = __builtin_amdgcn_wmma_f32_16x16x4_f32(
                    false, a0, false, b1, (short)0, acc01, false, false);
        acc10 = __builtin_amdgcn_wmma_f32_16x16x4_f32(
                    false, a1, false, b0, (short)0, acc10, false, false);
        acc11 = __builtin_amdgcn_wmma_f32_16x16x4_f32(
                    false, a1, false, b1, (short)0, acc11, false, false);

        a0 = na0; a1 = na1; b0 = nb0; b1 = nb1;
    }
    // epilogue: last k-step
    acc00 = __builtin_amdgcn_wmma_f32_16x16x4_f32(
                false, a0, false, b0, (short)0, acc00, false, false);
    acc01 = __builtin_amdgcn_wmma_f32_16x16x4_f32(
                false, a0, false, b1, (short)0, acc01, false, false);
    acc10 = __builtin_amdgcn_wmma_f32_16x16x4_f32(
                false, a1, false, b0, (short)0, acc10, false, false);
    acc11 = __builtin_amdgcn_wmma_f32_16x16x4_f32(
                false, a1, false, b1, (short)0, acc11, false, false);

    float bv0 = 0.0f, bv1 = 0.0f;
    if (bias) { bv0 = bias[n0 + ln]; bv1 = bias[n0 + 16 + ln]; }

#pragma unroll
    for (int j = 0; j < 8; ++j) {
        const int row = m0 + j + 8 * kh;
        float* cp = C + (size_t)row * N + n0 + ln;
        cp[0]  = acc00[j] + bv0;
        cp[16] = acc01[j] + bv1;
        float* cp2 = C + (size_t)(row + 16) * N + n0 + ln;
        cp2[0]  = acc10[j] + bv0;
        cp2[16] = acc11[j] + bv1;
    }
}

// =====================================================================
// Recurrent scan: one workgroup per (b,h). kv state [FD,DH] = 64 KB lives
// in LDS for all 64 timesteps; only the final state is written to HBM.
// 256 threads = 8 wave32s. Two phases per step:
//   Phase A (fd = tid): nm update + denominator partial (wave32 shfl reduce)
//   Phase B (d = tid&63, g = tid>>6): kv rank-1 discounted update + numerator
// =====================================================================
__global__ void __launch_bounds__(256)
galite_scan(const float* __restrict__ Qp,  const float* __restrict__ Kp,
            const float* __restrict__ Vp,  const float* __restrict__ Bpre,
            const float* __restrict__ Gpre,
            const float* __restrict__ P1,  const float* __restrict__ P2,
            const float* __restrict__ P3,
            const int*   __restrict__ term,
            const float* __restrict__ kv0, const float* __restrict__ nm0,
            float* __restrict__ attn_out,
            float* __restrict__ kv_last,   float* __restrict__ nm_last) {
    __shared__ float kv_s[FD * DH];     // 64 KB resident state
    __shared__ float nm_s[FD];
    __shared__ float q_s[DH], k_s[DH], gv_s[DH], gs_s[DH];
    __shared__ float rp1[ETA_], rp2[ETA_], sp3[ETA_];
    __shared__ float red[256];
    __shared__ float wred[8];
    __shared__ float maskS;

    const int tid  = threadIdx.x;
    const int bh   = blockIdx.x;        // b*HEADS + h
    const int b    = bh / HEADS;
    const int h    = bh % HEADS;
    const int d    = tid & 63;
    const int g    = tid >> 6;          // eta index for phase B
    const int lane = tid & 31;
    const int wid  = tid >> 5;

    // init state from inputs
    for (int i = tid; i < FD * DH; i += 256)
        kv_s[i] = kv0[(size_t)bh * FD * DH + i];
    nm_s[tid] = nm0[(size_t)bh * FD + tid];

    for (int t = 0; t < T_STEPS; ++t) {
        __syncthreads();
        const int row = t * BATCH + b;
        if (tid < DH) {
            const size_t ofs = (size_t)row * DIM + h * DH + tid;
            q_s[tid]  = Qp[ofs];
            k_s[tid]  = Kp[ofs];
            gv_s[tid] = Vp[ofs] * sigmoidf_(Bpre[ofs]);   // v * sigmoid(beta)
            gs_s[tid] = sigmoidf_(Gpre[ofs]);             // sigmoid(gamma)
        } else if (tid < DH + ETA_) {
            const int e = tid - DH;
            rp1[e] = fmaxf(P1[(size_t)row * (HEADS * ETA_) + h * ETA_ + e], 0.0f);
        } else if (tid < DH + 2 * ETA_) {
            const int e = tid - DH - ETA_;
            rp2[e] = fmaxf(P2[(size_t)row * (HEADS * ETA_) + h * ETA_ + e], 0.0f);
        } else if (tid < DH + 3 * ETA_) {
            const int e = tid - DH - 2 * ETA_;
            sp3[e] = sigmoidf_(P3[(size_t)row * (HEADS * ETA_) + h * ETA_ + e]);
        } else if (tid == DH + 3 * ETA_) {
            maskS = 1.0f - (float)term[t * BATCH + b];
        }
        __syncthreads();

        // ---- Phase A: fd = tid. nm update + denominator partial ----
        {
            const int e  = tid >> 6;
            const int dd = tid & 63;
            const float gf   = sp3[e] * gs_s[dd];            // gamma_feat
            const float gk   = rp1[e] * k_s[dd] * gf;        // gated_k
            const float disc = (1.0f - gf) * maskS;
            const float nmv  = disc * nm_s[tid] + gk;
            nm_s[tid] = nmv;
            float den = (rp2[e] * q_s[dd]) * nmv;            // phi_q * nm
            for (int off = 16; off > 0; off >>= 1)
                den += __shfl_down(den, off, 32);
            if (lane == 0) wred[wid] = den;
        }

        // ---- Phase B: kv update + numerator accumulation ----
        {
            float num = 0.0f;
            const float gvd  = gv_s[d];
            const float r1g  = rp1[g];
            const float r2g  = rp2[g];
            const float s3g  = sp3[g];
#pragma unroll 4
            for (int j = 0; j < DH; ++j) {
                const float gf   = s3g * gs_s[j];
                const float gk   = r1g * k_s[j] * gf;
                const float disc = (1.0f - gf) * maskS;
                const int   idx  = (g * DH + j) * DH + d;    // bank-conflict free
                const float kvv  = disc * kv_s[idx] + gvd * gk;
                kv_s[idx] = kvv;
                num += kvv * (r2g * q_s[j]);                 // * phi_q
            }
            red[tid] = num;
        }
        __syncthreads();

        if (tid < DH) {
            const float num = red[tid] + red[tid + 64] + red[tid + 128] + red[tid + 192];
            const float den = wred[0] + wred[1] + wred[2] + wred[3]
                            + wred[4] + wred[5] + wred[6] + wred[7];
            attn_out[(size_t)row * DIM + h * DH + tid] = num / (den + EPS_);
        }
    }

    __syncthreads();
    for (int i = tid; i < FD * DH; i += 256)
        kv_last[(size_t)bh * FD * DH + i] = kv_s[i];
    nm_last[(size_t)bh * FD + tid] = nm_s[tid];
}

// =====================================================================
extern "C" void kernel_launch(void* const* d_in, const int* in_sizes, int n_in,
                              void* d_out, int out_size, void* d_ws, size_t ws_size,
                              hipStream_t stream) {
    const float* inputs = (const float*)d_in[0];
    const int*   term   = (const int*)  d_in[1];
    const float* kv0    = (const float*)d_in[2];
    const float* nm0    = (const float*)d_in[3];
    const float* Wq     = (const float*)d_in[4];
    const float* Wk     = (const float*)d_in[5];
    const float* Wv     = (const float*)d_in[6];
    const float* Wb     = (const float*)d_in[7];
    const float* Wg     = (const float*)d_in[8];
    const float* Wp1    = (const float*)d_in[9];
    const float* Wp2    = (const float*)d_in[10];
    const float* Wp3    = (const float*)d_in[11];
    const float* Wo     = (const float*)d_in[12];
    const float* bo     = (const float*)d_in[13];

    float* out = (float*)d_out;
    float* ws  = (float*)d_ws;

    // workspace layout (floats)
    const size_t BIG = (size_t)ROWS * DIM;            // 524288
    const size_t SML = (size_t)ROWS * (HEADS * ETA_); // 32768
    float* Qp   = ws;
    float* Kp   = Qp + BIG;
    float* Vp   = Kp + BIG;
    float* Bp   = Vp + BIG;
    float* Gp   = Bp + BIG;
    float* P1   = Gp + BIG;
    float* P2   = P1 + SML;
    float* P3   = P2 + SML;
    float* AOut = P3 + SML;

    // output layout: output [T,B,DIM] | kv_last [B,H,FD,DH] | nm_last [B,H,FD]
    float* out_main = out;
    float* kv_last  = out + BIG;
    float* nm_last  = kv_last + (size_t)BATCH * HEADS * FD * DH;

    auto gemm = [&](const float* X, const float* Wm, const float* bias,
                    float* C, int N) {
        const int jobs   = (ROWS / 32) * (N / 32);
        const int blocks = (jobs + 7) / 8;           // 8 waves / block
        gemm_xwT_wmma<<<blocks, 256, 0, stream>>>(X, Wm, bias, C, ROWS, N, DIM);
    };

    // Phase 1: projections (WMMA f32)
    gemm(inputs, Wq,  nullptr, Qp, DIM);
    gemm(inputs, Wk,  nullptr, Kp, DIM);
    gemm(inputs, Wv,  nullptr, Vp, DIM);
    gemm(inputs, Wb,  nullptr, Bp, DIM);
    gemm(inputs, Wg,  nullptr, Gp, DIM);
    gemm(inputs, Wp1, nullptr, P1, HEADS * ETA_);
    gemm(inputs, Wp2, nullptr, P2, HEADS * ETA_);
    gemm(inputs, Wp3, nullptr, P3, HEADS * ETA_);

    // Phase 2: LDS-resident recurrent scan, one WG per (b,h)
    galite_scan<<<BATCH * HEADS, 256, 0, stream>>>(
        Qp, Kp, Vp, Bp, Gp, P1, P2, P3, term, kv0, nm0,
        AOut, kv_last, nm_last);

    // Phase 3: output projection with bias (WMMA f32)
    gemm(AOut, Wo, bo, out_main, DIM);
}